// GraphEncoder_17952963298146
// MI455X (gfx1250) — compile-verified
//
#include <hip/hip_runtime.h>
#include <hip/hip_bf16.h>

typedef __attribute__((ext_vector_type(2))) float v2f;
typedef __attribute__((ext_vector_type(8))) float v8f;

#define BGR   32                 // graphs
#define NNODE 2000               // nodes / graph
#define NEDGE 32000              // edges / graph
#define DIM   128                // embedding dim
#define NREL  39                 // relations
#define NBASE 8                  // bases
#define NRTOT (NNODE * NREL)     // 78000 (dst,rel) buckets per graph
#define SROW  ((NBASE + 1) * DIM) // 1152 = [S_x | S_z0..S_z7]

#define CNT_WORDS (BGR * NRTOT)  // int counts
#define S_WORDS   (BGR * SROW)   // float accumulators

// ---------------------------------------------------------------- zero ws
__global__ void k_zero(unsigned int* __restrict__ ws) {
    int i = blockIdx.x * blockDim.x + threadIdx.x;
    if (i < CNT_WORDS + S_WORDS) ws[i] = 0u;
}

// ------------------------------------------------- per-(dst,rel) counts
__global__ void k_count(const int* __restrict__ edge_index,
                        const int* __restrict__ edge_type,
                        int* __restrict__ cnt) {
    const int BLK_PER_G = NEDGE / 256;              // 125
    int b = blockIdx.x / BLK_PER_G;
    int e = (blockIdx.x % BLK_PER_G) * 256 + threadIdx.x;
    int dst = edge_index[(b * 2 + 1) * NEDGE + e];
    int et  = edge_type[b * NEDGE + e];
    atomicAdd(&cnt[b * NRTOT + dst * NREL + et], 1);
}

// --------------------------------------------- S_x = sum_n emb[node_ids]
__global__ void k_nodesum(const int* __restrict__ node_ids,
                          const float* __restrict__ emb,
                          float* __restrict__ S) {
    const int BLK_PER_G = 4;                        // 500 nodes per block
    int b     = blockIdx.x / BLK_PER_G;
    int chunk = blockIdx.x % BLK_PER_G;
    int d   = threadIdx.x & (DIM - 1);
    int sub = threadIdx.x >> 7;                     // 0/1 (2 nodes in flight)
    int n0 = chunk * 500;
    float acc = 0.f;
    for (int n = n0 + sub; n < n0 + 500; n += 2) {
        int id = node_ids[b * NNODE + n];
        acc += emb[id * DIM + d];
    }
    __shared__ float red[DIM];
    if (sub == 1) red[d] = acc;
    __syncthreads();
    if (sub == 0) atomicAdd(&S[b * SROW + d], acc + red[d]);
}

// -------------------- S_zk = sum_e (comp[et,k]/cnt[dst,et]) * emb[src]
__global__ void k_edgeacc(const int* __restrict__ edge_index,
                          const int* __restrict__ edge_type,
                          const float* __restrict__ emb,
                          const float* __restrict__ comp,
                          const int* __restrict__ cnt,
                          float* __restrict__ S) {
    const int BLK_PER_G = 20;                       // 1600 edges per block
    const int EPB  = NEDGE / BLK_PER_G;             // 1600
    const int TILE = 32;                            // edges staged in LDS
    int b  = blockIdx.x / BLK_PER_G;
    int e0 = (blockIdx.x % BLK_PER_G) * EPB;
    int t   = threadIdx.x;
    int d   = t & (DIM - 1);
    int sub = t >> 7;

    __shared__ int   lds_src[TILE];
    __shared__ float lds_w[TILE * NBASE];
    __shared__ float red[NBASE * DIM];

    float acc[NBASE];
#pragma unroll
    for (int k = 0; k < NBASE; ++k) acc[k] = 0.f;

    const int* src_arr = edge_index + (b * 2 + 0) * NEDGE;
    const int* dst_arr = edge_index + (b * 2 + 1) * NEDGE;
    const int* et_arr  = edge_type + b * NEDGE;
    const int* cnt_b   = cnt + b * NRTOT;

    for (int tb = 0; tb < EPB; tb += TILE) {
        __syncthreads();
        if (t < TILE) {                              // stage per-edge scalars
            int e   = e0 + tb + t;
            int src = src_arr[e];
            int dst = dst_arr[e];
            int et  = et_arr[e];
            float inv = 1.0f / (float)cnt_b[dst * NREL + et];
            lds_src[t] = src;
#pragma unroll
            for (int k = 0; k < NBASE; ++k)
                lds_w[t * NBASE + k] = comp[et * NBASE + k] * inv;
        }
        __syncthreads();
        for (int i = sub; i < TILE; i += 2) {        // coalesced 512B row gather
            float x = emb[lds_src[i] * DIM + d];
#pragma unroll
            for (int k = 0; k < NBASE; ++k)
                acc[k] += lds_w[i * NBASE + k] * x;
        }
    }
    __syncthreads();
    if (sub == 1) {
#pragma unroll
        for (int k = 0; k < NBASE; ++k) red[k * DIM + d] = acc[k];
    }
    __syncthreads();
    if (sub == 0) {
#pragma unroll
        for (int k = 0; k < NBASE; ++k)
            atomicAdd(&S[b * SROW + DIM + k * DIM + d], acc[k] + red[k * DIM + d]);
    }
}

// ---- G[32,128] = S[32,1152] @ [root; bases][1152,128] + N*bias, L2-norm
__global__ void __launch_bounds__(512)
k_wmma_out(const float* __restrict__ S, const float* __restrict__ bases,
           const float* __restrict__ root, const float* __restrict__ bias,
           float* __restrict__ out) {
    int t     = threadIdx.x;
    int wave  = t >> 5;           // 16 waves = 2 Mtiles x 8 Ntiles
    int lane  = t & 31;
    int mt    = wave >> 3;        // 0..1
    int nt    = wave & 7;         // 0..7
    int m     = lane & 15;
    int khalf = lane >> 4;        // which pair of K values this half-wave holds
    int row   = mt * 16 + m;      // graph id (A-matrix row)
    int col   = nt * 16 + (lane & 15);  // output channel (B/D column)

    v8f acc = {};
    for (int kk = 0; kk < SROW; kk += 4) {
        int ka = kk + khalf * 2;  // even, never straddles the root/bases split
        v2f a, bb;
        a.x = S[row * SROW + ka];
        a.y = S[row * SROW + ka + 1];
        const float* wrow = (ka < DIM) ? (root + ka * DIM)
                                       : (bases + (ka - DIM) * DIM);
        bb.x = wrow[col];
        bb.y = wrow[DIM + col];
        // D = A(16x4 f32) * B(4x16 f32) + C  -> v_wmma_f32_16x16x4_f32
        acc = __builtin_amdgcn_wmma_f32_16x16x4_f32(
            false, a, false, bb, (short)0, acc, false, false);
    }

    __shared__ float G[BGR * DIM];   // 16 KB
    __shared__ float invn[BGR];
    float bi = (float)NNODE * bias[col];
#pragma unroll
    for (int j = 0; j < 8; ++j) {    // D layout: M = j + khalf*8, N = lane&15
        int g = mt * 16 + khalf * 8 + j;
        G[g * DIM + col] = acc[j] + bi;
    }
    __syncthreads();
    if (t < BGR) {
        float s = 0.f;
        for (int d2 = 0; d2 < DIM; ++d2) { float v = G[t * DIM + d2]; s += v * v; }
        invn[t] = 1.0f / fmaxf(sqrtf(s), 1e-5f);
    }
    __syncthreads();
    for (int i = t; i < BGR * DIM; i += 512)
        out[i] = G[i] * invn[i / DIM];
}

extern "C" void kernel_launch(void* const* d_in, const int* in_sizes, int n_in,
                              void* d_out, int out_size, void* d_ws, size_t ws_size,
                              hipStream_t stream) {
    const int*   node_ids   = (const int*)d_in[0];
    const int*   edge_index = (const int*)d_in[1];
    const int*   edge_type  = (const int*)d_in[2];
    const float* embedding  = (const float*)d_in[3];
    const float* bases      = (const float*)d_in[4];
    const float* comp       = (const float*)d_in[5];
    const float* root       = (const float*)d_in[6];
    const float* bias       = (const float*)d_in[7];
    float* out = (float*)d_out;

    int*   cnt = (int*)d_ws;                       // [B * N * R] counts
    float* S   = (float*)d_ws + CNT_WORDS;         // [B * 1152] accumulators

    int total = CNT_WORDS + S_WORDS;
    k_zero   <<<(total + 255) / 256, 256, 0, stream>>>((unsigned int*)d_ws);
    k_count  <<<BGR * (NEDGE / 256), 256, 0, stream>>>(edge_index, edge_type, cnt);
    k_nodesum<<<BGR * 4,             256, 0, stream>>>(node_ids, embedding, S);
    k_edgeacc<<<BGR * 20,            256, 0, stream>>>(edge_index, edge_type,
                                                       embedding, comp, cnt, S);
    k_wmma_out<<<1, 512, 0, stream>>>(S, bases, root, bias, out);
}